// DynamicsFunctions_59167469469780
// MI455X (gfx1250) — compile-verified
//
#include <hip/hip_runtime.h>
#include <cmath>

typedef __attribute__((ext_vector_type(2))) float v2f;
typedef __attribute__((ext_vector_type(4))) float v4f;
typedef __attribute__((ext_vector_type(8))) float v8f;
typedef __attribute__((ext_vector_type(4))) int   v4i;

#define DENSITY_C 1000.0f
#define DT_C      0.01f
#define BETA_C    0.25f
#define MU_C      1.0f
#define LAM_C     1.0f
// rho / (beta * dt^2) = 4e7
#define CM_C      (DENSITY_C / (BETA_C * DT_C * DT_C))

// One wave per element; 12x12 Newmark Hessian = rank-22 outer-product sum,
// evaluated with 6 x V_WMMA_F32_16X16X4_F32 into a 16x16 f32 accumulator tile.
// All rank-factor columns are individually named SSA floats so clang cannot
// re-merge them into a dynamically indexed aggregate (which it lowers into
// huge v_cmp/v_cndmask chains).
__global__ __launch_bounds__(256) void newmark_hess_kernel(
    const float* __restrict__ U, const float* __restrict__ Up,
    const int*   __restrict__ conns, const float* __restrict__ shapes,
    const float* __restrict__ sg, const float* __restrict__ vols,
    float* __restrict__ out, int E)
{
  const int lane = (int)(threadIdx.x & 31u);
  const int e = (int)(blockIdx.x * (blockDim.x >> 5) + (threadIdx.x >> 5));
  if (e >= E) return;               // wave-uniform: EXEC stays all-1 for WMMA

  __builtin_prefetch(sg + (size_t)(e + 16) * 12, 0, 3);  // global_prefetch_b8

  // ---------- uniform (wave-replicated) element setup ----------
  const v4i nd = *(const v4i*)(conns + (size_t)e * 4);       // 16B aligned
  const float* sge = sg + (size_t)e * 12;                    // 48B stride, 16B aligned
  const v4f g0 = *(const v4f*)(sge);
  const v4f g1 = *(const v4f*)(sge + 4);
  const v4f g2 = *(const v4f*)(sge + 8);
  const float Gf[12] = { g0.x, g0.y, g0.z, g0.w,
                         g1.x, g1.y, g1.z, g1.w,
                         g2.x, g2.y, g2.z, g2.w };
  float Wf[12];
#pragma unroll
  for (int n = 0; n < 4; ++n) {
    const int node = nd[n];
#pragma unroll
    for (int i = 0; i < 3; ++i)
      Wf[3 * n + i] = U[node * 3 + i] - Up[node * 3 + i];
  }
  const float vol = vols[e];

  // F = I + sum_n W[n] (x) G[n]   (constant indices only)
  float F[3][3];
#pragma unroll
  for (int i = 0; i < 3; ++i)
#pragma unroll
    for (int j = 0; j < 3; ++j) {
      float a = (i == j) ? 1.0f : 0.0f;
#pragma unroll
      for (int n = 0; n < 4; ++n) a += Wf[3 * n + i] * Gf[3 * n + j];
      F[i][j] = a;
    }

  const float J =
      F[0][0] * (F[1][1] * F[2][2] - F[1][2] * F[2][1]) -
      F[0][1] * (F[1][0] * F[2][2] - F[1][2] * F[2][0]) +
      F[0][2] * (F[1][0] * F[2][1] - F[1][1] * F[2][0]);
  const float invJ = 1.0f / J;
  const float lJ = logf(J);

  float Fi[3][3];
  Fi[0][0] = (F[1][1] * F[2][2] - F[1][2] * F[2][1]) * invJ;
  Fi[0][1] = (F[0][2] * F[2][1] - F[0][1] * F[2][2]) * invJ;
  Fi[0][2] = (F[0][1] * F[1][2] - F[0][2] * F[1][1]) * invJ;
  Fi[1][0] = (F[1][2] * F[2][0] - F[1][0] * F[2][2]) * invJ;
  Fi[1][1] = (F[0][0] * F[2][2] - F[0][2] * F[2][0]) * invJ;
  Fi[1][2] = (F[0][2] * F[1][0] - F[0][0] * F[1][2]) * invJ;
  Fi[2][0] = (F[1][0] * F[2][1] - F[1][1] * F[2][0]) * invJ;
  Fi[2][1] = (F[0][1] * F[2][0] - F[0][0] * F[2][1]) * invJ;
  Fi[2][2] = (F[0][0] * F[1][1] - F[0][1] * F[1][0]) * invJ;

  const float c1  = vol * LAM_C;                 // lambda b b^T
  const float c2  = vol * (MU_C - LAM_C * lJ);   // swap term
  const float cmu = vol * MU_C;                  // mu g_{nm} delta_{ik}
  const float ccm = vol * CM_C;                  // mass term

  // ---------- per-lane ingredients ----------
  const int  r    = lane & 15;          // row (n,i) for A == col (m,k) for B
  const int  nn   = r / 3;
  const int  ii   = r - 3 * nn;
  const bool hi   = lane >= 16;
  const bool vrow = (r < 12);
  const int  nnc  = vrow ? nn : 3;      // clamp so pad rows don't read OOB

  // lane's own G-row and shape value (3 lanes share an address -> L0 hits)
  const float* gr = sge + 3 * nnc;
  const float Gr0 = gr[0], Gr1 = gr[1], Gr2 = gr[2];
  const float sn  = shapes[e * 4 + nnc];

  // lane's own b-row: b[n]_q = sum_j Fi[j][q] * G[n][j]
  const float br0 = Fi[0][0] * Gr0 + Fi[1][0] * Gr1 + Fi[2][0] * Gr2;
  const float br1 = Fi[0][1] * Gr0 + Fi[1][1] * Gr1 + Fi[2][1] * Gr2;
  const float br2 = Fi[0][2] * Gr0 + Fi[1][2] * Gr1 + Fi[2][2] * Gr2;
  const float bii = (ii == 0) ? br0 : ((ii == 1) ? br1 : br2);

  // float masks: m[p] = (ii == p && row valid) ? 1 : 0
  const float vf = vrow ? 1.0f : 0.0f;
  const float m0 = (vrow && ii == 0) ? 1.0f : 0.0f;
  const float m1 = (vrow && ii == 1) ? 1.0f : 0.0f;
  const float m2 = (vrow && ii == 2) ? 1.0f : 0.0f;

  const float c2b0 = c2 * br0, c2b1 = c2 * br1, c2b2 = c2 * br2;
  const float cg0 = cmu * Gr0, cg1 = cmu * Gr1, cg2 = cmu * Gr2;
  const float cs  = ccm * sn;

  // ---------- 22 rank-factor columns as NAMED scalars ----------
  // k = 0 : lambda * b (x) b
  const float fa0 = c1 * bii * vf,  fb0 = bii * vf;
  // k = 1..9 : (mu - lam*lJ) b[n]_q d_{ip}  (x)  b[m]_p d_{kq}
  const float fa1 = c2b0 * m0, fb1 = br0 * m0;
  const float fa2 = c2b1 * m0, fb2 = br0 * m1;
  const float fa3 = c2b2 * m0, fb3 = br0 * m2;
  const float fa4 = c2b0 * m1, fb4 = br1 * m0;
  const float fa5 = c2b1 * m1, fb5 = br1 * m1;
  const float fa6 = c2b2 * m1, fb6 = br1 * m2;
  const float fa7 = c2b0 * m2, fb7 = br2 * m0;
  const float fa8 = c2b1 * m2, fb8 = br2 * m1;
  const float fa9 = c2b2 * m2, fb9 = br2 * m2;
  // k = 10..18 : mu G[n,j] d_{ip}  (x)  G[m,j] d_{kp}
  const float fa10 = cg0 * m0, fb10 = Gr0 * m0;
  const float fa11 = cg0 * m1, fb11 = Gr0 * m1;
  const float fa12 = cg0 * m2, fb12 = Gr0 * m2;
  const float fa13 = cg1 * m0, fb13 = Gr1 * m0;
  const float fa14 = cg1 * m1, fb14 = Gr1 * m1;
  const float fa15 = cg1 * m2, fb15 = Gr1 * m2;
  const float fa16 = cg2 * m0, fb16 = Gr2 * m0;
  const float fa17 = cg2 * m1, fb17 = Gr2 * m1;
  const float fa18 = cg2 * m2, fb18 = Gr2 * m2;
  // k = 19..21 : Cm s_n d_{ip}  (x)  s_m d_{kp}
  const float fa19 = cs * m0, fb19 = sn * m0;
  const float fa20 = cs * m1, fb20 = sn * m1;
  const float fa21 = cs * m2, fb21 = sn * m2;

  // ---------- 6 x v_wmma_f32_16x16x4_f32 ----------
  // A 16x4 f32 layout: VGPR0 = K{0|2}, VGPR1 = K{1|3} (lo|hi half-wave).
  // Each operand: one v_cndmask between two named SSA values.
  v8f acc = {};
  v2f av, bv;

  av.x = hi ? fa2 : fa0;  av.y = hi ? fa3 : fa1;
  bv.x = hi ? fb2 : fb0;  bv.y = hi ? fb3 : fb1;
  acc = __builtin_amdgcn_wmma_f32_16x16x4_f32(false, av, false, bv, (short)0, acc, false, false);

  av.x = hi ? fa6 : fa4;  av.y = hi ? fa7 : fa5;
  bv.x = hi ? fb6 : fb4;  bv.y = hi ? fb7 : fb5;
  acc = __builtin_amdgcn_wmma_f32_16x16x4_f32(false, av, false, bv, (short)0, acc, false, false);

  av.x = hi ? fa10 : fa8;  av.y = hi ? fa11 : fa9;
  bv.x = hi ? fb10 : fb8;  bv.y = hi ? fb11 : fb9;
  acc = __builtin_amdgcn_wmma_f32_16x16x4_f32(false, av, false, bv, (short)0, acc, false, false);

  av.x = hi ? fa14 : fa12;  av.y = hi ? fa15 : fa13;
  bv.x = hi ? fb14 : fb12;  bv.y = hi ? fb15 : fb13;
  acc = __builtin_amdgcn_wmma_f32_16x16x4_f32(false, av, false, bv, (short)0, acc, false, false);

  av.x = hi ? fa18 : fa16;  av.y = hi ? fa19 : fa17;
  bv.x = hi ? fb18 : fb16;  bv.y = hi ? fb19 : fb17;
  acc = __builtin_amdgcn_wmma_f32_16x16x4_f32(false, av, false, bv, (short)0, acc, false, false);

  av.x = hi ? 0.0f : fa20;  av.y = hi ? 0.0f : fa21;   // cols 22/23 are zero pad
  bv.x = hi ? 0.0f : fb20;  bv.y = hi ? 0.0f : fb21;
  acc = __builtin_amdgcn_wmma_f32_16x16x4_f32(false, av, false, bv, (short)0, acc, false, false);

  // ---------- store 12x12 corner of the 16x16 tile ----------
  // D layout: VGPR rr = row rr (lanes 0-15) / row rr+8 (lanes 16-31), col = lane&15.
  const size_t base = (size_t)e * 144 + (size_t)r + (hi ? 96u : 0u);
#pragma unroll
  for (int rr = 0; rr < 8; ++rr) {
    const bool pred = vrow && (rr < 4 || !hi);   // row<12 && col<12
    if (pred) out[base + (size_t)(rr * 12)] = acc[rr];
  }
}

extern "C" void kernel_launch(void* const* d_in, const int* in_sizes, int n_in,
                              void* d_out, int out_size, void* d_ws, size_t ws_size,
                              hipStream_t stream) {
  const float* U      = (const float*)d_in[0];
  const float* Up     = (const float*)d_in[1];
  // d_in[2] = state (unused)
  const int*   conns  = (const int*)d_in[3];
  const float* shapes = (const float*)d_in[4];
  const float* sg     = (const float*)d_in[5];
  const float* vols   = (const float*)d_in[6];
  float* out = (float*)d_out;

  const int E = in_sizes[3] / 4;        // conns is [E,4]
  const int wavesPerBlock = 8;          // 256 threads = 8 wave32
  const int blocks = (E + wavesPerBlock - 1) / wavesPerBlock;
  newmark_hess_kernel<<<blocks, 256, 0, stream>>>(U, Up, conns, shapes, sg, vols, out, E);
}